// GCN_19911468384339
// MI455X (gfx1250) — compile-verified
//
#include <hip/hip_runtime.h>

typedef __attribute__((ext_vector_type(2))) float v2f;
typedef __attribute__((ext_vector_type(8))) float v8f;

#define IN_F  128
#define OUT_F 64

// ---------------------------------------------------------------------------
// SpMM: out[d] += vals[e] * x[src[e]]   (one wave32 per edge, float4 per lane,
// 4x global_atomic_add_f32 per lane -> 128-float row per wave)
// ---------------------------------------------------------------------------
__global__ void gcn_spmm_kernel(const float* __restrict__ x,
                                const float* __restrict__ vals,
                                const int* __restrict__ src,
                                const int* __restrict__ dst,
                                float* __restrict__ out, int nEdges)
{
    int wave = blockIdx.x * (blockDim.x >> 5) + (threadIdx.x >> 5);
    if (wave >= nEdges) return;
    int lane = threadIdx.x & 31;

    int s = src[wave];
    int d = dst[wave];
    float v = vals[wave];

    const float4 xa = *((const float4*)(x + (size_t)s * IN_F) + lane);
    float* o = out + (size_t)d * IN_F + lane * 4;
    atomicAdd(o + 0, v * xa.x);
    atomicAdd(o + 1, v * xa.y);
    atomicAdd(o + 2, v * xa.z);
    atomicAdd(o + 3, v * xa.w);
}

// ---------------------------------------------------------------------------
// Dense layer: Y[m][o] = sum_k A[m][k] * W[o][k]   (y = A @ W^T, torch Linear)
// fp32 WMMA 16x16x4, one wave per 16x16 output tile, K=128 -> 32 WMMAs/tile.
// Fragment layout (f32, wave32): lane-half h=lane/16, VGPR v  =>  K = v + 2h,
// so a float2 load at (k + 2h) fills both A-frag VGPRs; same pattern for B
// because B[k][o] = W[o][k] is read along W's row o.
// C/D layout: VGPR i holds row (i + 8h), col lane%16.
// ---------------------------------------------------------------------------
__global__ void gcn_gemm_wmma_f32(const float* __restrict__ A,
                                  const float* __restrict__ W,
                                  float* __restrict__ Y,
                                  int ldy, int relu)
{
    const int lane   = threadIdx.x;      // 0..31, one full wave, EXEC all-1s
    const int half   = lane >> 4;        // 0 or 1
    const int ln16   = lane & 15;
    const int rowTile = blockIdx.x << 4;
    const int colTile = blockIdx.y << 4;

    const float* arow = A + (size_t)(rowTile + ln16) * IN_F + 2 * half;
    const float* brow = W + (size_t)(colTile + ln16) * IN_F + 2 * half;

    v8f acc = {};
#pragma unroll
    for (int k = 0; k < IN_F; k += 4) {
        v2f a = *(const v2f*)(arow + k);
        v2f b = *(const v2f*)(brow + k);
        acc = __builtin_amdgcn_wmma_f32_16x16x4_f32(
            /*neg_a=*/false, a, /*neg_b=*/false, b,
            /*c_mod=*/(short)0, acc, /*reuse_a=*/false, /*reuse_b=*/false);
    }

    float* yrow = Y + (size_t)(rowTile + 8 * half) * ldy + colTile + ln16;
#pragma unroll
    for (int i = 0; i < 8; ++i) {
        float r = acc[i];
        if (relu) r = fmaxf(r, 0.0f);
        yrow[(size_t)i * ldy] = r;
    }
}

// ---------------------------------------------------------------------------
// Row softmax over 64 classes: one wave32 per row, 2 elements per lane,
// wave32 shuffle reductions (warpSize==32 on gfx1250).
// ---------------------------------------------------------------------------
__global__ void gcn_softmax_kernel(const float* __restrict__ logits,
                                   float* __restrict__ out, int nRows)
{
    int row = blockIdx.x * (blockDim.x >> 5) + (threadIdx.x >> 5);
    if (row >= nRows) return;
    int lane = threadIdx.x & 31;

    const float* lr = logits + (size_t)row * OUT_F;
    float a = lr[lane];
    float b = lr[lane + 32];

    float m = fmaxf(a, b);
#pragma unroll
    for (int off = 16; off > 0; off >>= 1)
        m = fmaxf(m, __shfl_xor(m, off, 32));

    float e0 = __expf(a - m);
    float e1 = __expf(b - m);
    float s = e0 + e1;
#pragma unroll
    for (int off = 16; off > 0; off >>= 1)
        s += __shfl_xor(s, off, 32);

    float inv = 1.0f / s;
    float* orow = out + (size_t)row * OUT_F;
    orow[lane]      = e0 * inv;
    orow[lane + 32] = e1 * inv;
}

// ---------------------------------------------------------------------------
// inputs: 0=x [N,128] f32, 1=vals [E] f32, 2=W1 [128,128] f32,
//         3=W2 [64,128] f32, 4=src [E] i32, 5=dst [E] i32
// output: softmax probs [N,64] f32
// ---------------------------------------------------------------------------
extern "C" void kernel_launch(void* const* d_in, const int* in_sizes, int n_in,
                              void* d_out, int out_size, void* d_ws, size_t ws_size,
                              hipStream_t stream)
{
    const float* x    = (const float*)d_in[0];
    const float* vals = (const float*)d_in[1];
    const float* W1   = (const float*)d_in[2];
    const float* W2   = (const float*)d_in[3];
    const int*   src  = (const int*)d_in[4];
    const int*   dst  = (const int*)d_in[5];

    const int nE = in_sizes[1];
    const int nN = in_sizes[0] / IN_F;   // 50000

    // workspace layout (fp32): agg1 | h | agg2 | logits   (~89.6 MB total)
    float* agg1   = (float*)d_ws;
    float* h      = agg1 + (size_t)nN * IN_F;
    float* agg2   = h    + (size_t)nN * IN_F;
    float* logits = agg2 + (size_t)nN * IN_F;

    // zero the scatter-add accumulators (capture-safe memset nodes)
    hipMemsetAsync(agg1, 0, (size_t)nN * IN_F * sizeof(float), stream);
    hipMemsetAsync(agg2, 0, (size_t)nN * IN_F * sizeof(float), stream);

    const int wavesPerBlk = 8;           // 256 threads = 8 wave32
    const int spmmGrid = (nE + wavesPerBlk - 1) / wavesPerBlk;

    // layer 1: agg1 = A @ x ; h = relu(agg1 @ W1^T)
    gcn_spmm_kernel<<<spmmGrid, 256, 0, stream>>>(x, vals, src, dst, agg1, nE);
    gcn_gemm_wmma_f32<<<dim3(nN / 16, IN_F / 16), 32, 0, stream>>>(agg1, W1, h, IN_F, 1);

    // layer 2: agg2 = A @ h ; logits = agg2 @ W2^T ; softmax
    gcn_spmm_kernel<<<spmmGrid, 256, 0, stream>>>(h, vals, src, dst, agg2, nE);
    gcn_gemm_wmma_f32<<<dim3(nN / 16, OUT_F / 16), 32, 0, stream>>>(agg2, W2, logits, OUT_F, 0);

    gcn_softmax_kernel<<<(nN + wavesPerBlk - 1) / wavesPerBlk, 256, 0, stream>>>(
        logits, (float*)d_out, nN);
}